// FullFloodFillNetwork_609885356697
// MI455X (gfx1250) — compile-verified
//
#include <hip/hip_runtime.h>
#include <hip/hip_bf16.h>

// ---------------------------------------------------------------------------
// FullFloodFillNetwork for MI455X (gfx1250, wave32)
//
// Phases:
//  1) hipMemcpyAsync: x -> d_out (reference returns x unchanged)
//  2) gemm_norm_kernel: y = W @ x via V_WMMA_F32_16X16X4_F32 (exact f32 path),
//     L2-normalize per head in LDS, store transposed qnT/knT [B,F,C] so the
//     flood-fill frontier gathers are coalesced.
//  3) ktot_kernel: deterministic sum of normalized k over F.
//  4) anchor_kernel: extract normalized k at anchor face.
//  5) scores0_kernel: initial scores from cos(q_f, k_anchor), LDS-tiled.
//  6) flood_kernel: 264-step BFS flood fill, one workgroup per batch,
//     all state (scores, done bits, frontier lists) in LDS. Per step only the
//     frontier's q/k columns are read (~L*2KB instead of 64MB).
// ---------------------------------------------------------------------------

typedef __attribute__((ext_vector_type(2))) float v2f;
typedef __attribute__((ext_vector_type(8))) float v8f;

#define BN 4
#define CC 256
#define FF 8192
#define HH 8
#define NSTEPS 264        // 4*GRID + 8
#define MAXFRONT 4096     // BFS frontier on 64x64 tri-grid stays well below this

// ---------------------------------------------------------------------------
// Kernel 2: WMMA GEMM + per-head L2 normalization, transposed store.
// Block = 128 threads (4 waves). Block covers one 16-wide f tile, full o=0..255,
// for one (batch, {q,k}). Wave w owns o in [64w, 64w+64) = 4 M-tiles.
// K loop: 64 iterations of K=4 f32 WMMA.
// ---------------------------------------------------------------------------
__global__ __launch_bounds__(128) void gemm_norm_kernel(
    const float* __restrict__ x, const float* __restrict__ Wq,
    const float* __restrict__ Wk,
    float* __restrict__ qnT, float* __restrict__ knT)
{
  const int f0  = blockIdx.x * 16;
  const int b   = blockIdx.y;
  const int isK = blockIdx.z;
  const float* W   = isK ? Wk : Wq;
  float*       out = isK ? knT : qnT;

  const int t     = threadIdx.x;
  const int wave  = t >> 5;
  const int lane  = t & 31;
  const int m     = lane & 15;          // M row inside tile / N col inside tile
  const int khalf = (lane >> 4) << 1;   // lanes 0-15 -> K+0,K+1 ; 16-31 -> K+2,K+3

  __shared__ float tile[16 * 257];      // [f 0..15][o 0..255], padded stride

  v8f acc[4] = {};
  const float* xb = x + (size_t)b * CC * FF;
  const int f = f0 + m;

  for (int kc = 0; kc < CC; kc += 4) {
    const int kb = kc + khalf;
    // B fragment (4x16 f32): column N = f, K = kb, kb+1
    v2f bf;
    bf[0] = xb[(size_t)kb * FF + f];
    bf[1] = xb[(size_t)(kb + 1) * FF + f];
#pragma unroll
    for (int mt = 0; mt < 4; ++mt) {
      const int o = wave * 64 + mt * 16 + m;
      const float2 wv = *(const float2*)(W + (size_t)o * CC + kb); // 8B aligned
      v2f af; af[0] = wv.x; af[1] = wv.y;
      acc[mt] = __builtin_amdgcn_wmma_f32_16x16x4_f32(
          /*neg_a=*/false, af, /*neg_b=*/false, bf,
          /*c_mod=*/(short)0, acc[mt], /*reuse_a=*/false, /*reuse_b=*/false);
    }
  }

  // Stage D tiles to LDS: VGPR i of lane l holds (M = i + 8*(l>=16), N = l&15)
  const int mhi = (lane >> 4) * 8;
#pragma unroll
  for (int mt = 0; mt < 4; ++mt) {
    const int obase = wave * 64 + mt * 16 + mhi;
#pragma unroll
    for (int i = 0; i < 8; ++i)
      tile[m * 257 + obase + i] = acc[mt][i];
  }
  __syncthreads();

  // Per-(f,head) L2 normalization: 16 f-rows * 8 heads = 128 threads
  {
    const int fr = t >> 3, h = t & 7;
    float* row = &tile[fr * 257 + h * 32];
    float ss = 0.f;
#pragma unroll
    for (int d = 0; d < 32; ++d) ss += row[d] * row[d];
    const float sc = rsqrtf(ss + 1e-12f);
#pragma unroll
    for (int d = 0; d < 32; ++d) row[d] *= sc;
  }
  __syncthreads();

  // Coalesced transposed store: out[b, f, c]
  float* ob = out + (size_t)b * FF * CC + (size_t)f0 * CC;
  for (int idx = t; idx < 16 * 256; idx += 128) {
    const int r = idx >> 8, c = idx & 255;
    ob[(size_t)r * CC + c] = tile[r * 257 + c];
  }
}

// ---------------------------------------------------------------------------
// Kernel 3: ktot[b,c] = sum_f knT[b,f,c]  (deterministic, no float atomics)
// ---------------------------------------------------------------------------
__global__ __launch_bounds__(256) void ktot_kernel(
    const float* __restrict__ knT, float* __restrict__ ktot)
{
  const int b = blockIdx.x, c = threadIdx.x;
  const float* base = knT + (size_t)b * FF * CC + c;
  float s0 = 0.f, s1 = 0.f, s2 = 0.f, s3 = 0.f;
  for (int f = 0; f < FF; f += 4) {
    s0 += base[(size_t)(f + 0) * CC];
    s1 += base[(size_t)(f + 1) * CC];
    s2 += base[(size_t)(f + 2) * CC];
    s3 += base[(size_t)(f + 3) * CC];
  }
  ktot[b * CC + c] = (s0 + s1) + (s2 + s3);
}

// ---------------------------------------------------------------------------
// Kernel 4: kanch[b,:] = knT[b, anchors[b], :]
// ---------------------------------------------------------------------------
__global__ void anchor_kernel(const float* __restrict__ knT,
                              const int* __restrict__ anchors,
                              float* __restrict__ kanch)
{
  const int b = blockIdx.x, t = threadIdx.x;
  const int a = anchors[b];
  kanch[b * CC + t] = knT[(size_t)b * FF * CC + (size_t)a * CC + t];
}

// ---------------------------------------------------------------------------
// Kernel 5: scores0[b,f] = (mean_h cos(q_f, k_anchor) + 1)/2
// Block = 256 threads over 32 f's, LDS-tiled for coalesced loads.
// ---------------------------------------------------------------------------
__global__ __launch_bounds__(256) void scores0_kernel(
    const float* __restrict__ qnT, const float* __restrict__ kanch,
    float* __restrict__ s0)
{
  __shared__ float ka[256];
  __shared__ float tile[32 * 257];
  const int b = blockIdx.y, f0 = blockIdx.x * 32, t = threadIdx.x;
  ka[t] = kanch[b * CC + t];
  const float* base = qnT + (size_t)b * FF * CC + (size_t)f0 * CC;
  for (int idx = t; idx < 32 * 256; idx += 256) {
    const int r = idx >> 8, c = idx & 255;
    tile[r * 257 + c] = base[idx];
  }
  __syncthreads();
  if (t < 32) {
    float acc = 0.f;
    const float* row = &tile[t * 257];
    for (int c = 0; c < 256; ++c) acc += row[c] * ka[c];
    s0[b * FF + f0 + t] = (acc * (1.0f / HH) + 1.0f) * 0.5f;
  }
}

// ---------------------------------------------------------------------------
// Kernel 6: flood fill. One workgroup (1024 threads) per batch; 264 sequential
// steps with all state in LDS. Frontier-sparse: per step we only touch the
// boundary faces' q/k columns (coalesced [f,C] rows).
// Static LDS total ~59 KB.
// ---------------------------------------------------------------------------
__global__ __launch_bounds__(1024) void flood_kernel(
    const float* __restrict__ qnT, const float* __restrict__ knT,
    const float* __restrict__ ktot, const float* __restrict__ s0,
    const int* __restrict__ adj, const int* __restrict__ anchors,
    float* __restrict__ out_scores)
{
  __shared__ float scores[FF];                    // 32 KB
  __shared__ unsigned short listA[MAXFRONT];      // 8 KB
  __shared__ unsigned short listB[MAXFRONT];      // 8 KB
  __shared__ unsigned int done_mask[FF / 32];     // 1 KB
  __shared__ unsigned int next_mask[FF / 32];     // 1 KB
  __shared__ float ktot_s[256];                   // 1 KB
  __shared__ float redq[1024], redk[1024];        // 8 KB
  __shared__ float pred_s;
  __shared__ int countA, countB;

  const int b  = blockIdx.x;
  const int t  = threadIdx.x;
  const int nt = blockDim.x; // 1024

  for (int f = t; f < FF; f += nt) scores[f] = s0[b * FF + f];
  for (int i = t; i < FF / 32; i += nt) done_mask[i] = 0u;
  if (t < 256) ktot_s[t] = ktot[b * CC + t];
  if (t == 0) {
    const int a = anchors[b];
    listA[0] = (unsigned short)a;
    countA = 1;
    done_mask[a >> 5] |= (1u << (a & 31));   // done0 includes the anchor
    countB = 0;
  }
  __syncthreads();

  const float* qb   = qnT + (size_t)b * FF * CC;
  const float* kbas = knT + (size_t)b * FF * CC;
  const int*   adjb = adj + (size_t)b * FF * 3;

  int parity = 0;
  for (int step = 0; step < NSTEPS; ++step) {
    unsigned short* cur = parity ? listB : listA;
    unsigned short* nxt = parity ? listA : listB;
    const int L = parity ? countB : countA;
    if (L == 0) break;

    // qs = sum_{f in boundary} q[:,f]; ksum likewise. c = channel, 4 slots.
    const int c = t & 255, slot = t >> 8;
    float pq = 0.f, pk = 0.f;
    for (int i = slot; i < L; i += 4) {
      const int f = cur[i];
      pq += qb[(size_t)f * CC + c];
      pk += kbas[(size_t)f * CC + c];
    }
    redq[t] = pq; redk[t] = pk;
    __syncthreads();
    if (t < 256) {
      const float qs = redq[t] + redq[t + 256] + redq[t + 512] + redq[t + 768];
      const float ks = redk[t] + redk[t + 256] + redk[t + 512] + redk[t + 768];
      redq[t] = qs * (ktot_s[t] - ks);   // per-channel dot term of qs . ko
    }
    __syncthreads();
    for (int s = 128; s > 0; s >>= 1) {
      if (t < s) redq[t] += redq[t + s];
      __syncthreads();
    }
    if (t == 0) {
      const float nb = (float)L, no = (float)(FF - L);
      const float mean_cos = redq[0] / ((float)HH * fmaxf(nb * no, 1.0f));
      pred_s = (mean_cos + 1.0f) * 0.5f;
      if (parity) countA = 0; else countB = 0;
    }
    for (int i = t; i < FF / 32; i += nt) next_mask[i] = 0u;
    __syncthreads();
    const float pred = pred_s;

    // Update scores on the boundary (reads done/scores of *done* neighbors only,
    // so in-place writes to boundary scores are race-free).
    for (int i = t; i < L; i += nt) {
      const int f = cur[i];
      float best = -3.402823466e38f;
      bool  any  = false;
#pragma unroll
      for (int j = 0; j < 3; ++j) {
        const int g = adjb[f * 3 + j];
        if (g >= 0 && ((done_mask[g >> 5] >> (g & 31)) & 1u)) {
          any = true;
          best = fmaxf(best, scores[g]);
        }
      }
      const float neigh = any ? best : 1.0f;
      scores[f] = fminf(fmaxf(pred, scores[f]), neigh);
    }
    __syncthreads();

    // done |= boundary
    for (int i = t; i < L; i += nt) {
      const int f = cur[i];
      atomicOr(&done_mask[f >> 5], 1u << (f & 31));
    }
    __syncthreads();

    // Expand frontier: neighbors of boundary, not done, deduped via next_mask.
    int* nxtCount = parity ? &countA : &countB;
    for (int i = t; i < L; i += nt) {
      const int f = cur[i];
#pragma unroll
      for (int j = 0; j < 3; ++j) {
        const int g = adjb[f * 3 + j];
        if (g >= 0 && !((done_mask[g >> 5] >> (g & 31)) & 1u)) {
          const unsigned int bit = 1u << (g & 31);
          const unsigned int old = atomicOr(&next_mask[g >> 5], bit);
          if (!(old & bit)) {
            const int idx = atomicAdd(nxtCount, 1);
            if (idx < MAXFRONT) nxt[idx] = (unsigned short)g;
          }
        }
      }
    }
    __syncthreads();
    parity ^= 1;
  }

  for (int f = t; f < FF; f += nt) out_scores[b * FF + f] = scores[f];
}

// ---------------------------------------------------------------------------
extern "C" void kernel_launch(void* const* d_in, const int* in_sizes, int n_in,
                              void* d_out, int out_size, void* d_ws, size_t ws_size,
                              hipStream_t stream)
{
  const float* x       = (const float*)d_in[0];
  const float* Wq      = (const float*)d_in[1];
  const float* Wk      = (const float*)d_in[2];
  const int*   adj     = (const int*)d_in[3];
  const int*   anchors = (const int*)d_in[4];

  float* out        = (float*)d_out;
  float* out_scores = out + (size_t)BN * CC * FF;

  // Workspace layout (floats): qnT[B*F*C] | knT[B*F*C] | ktot[B*256] |
  //                            kanch[B*256] | s0[B*F]   (~64.1 MB total)
  float* ws    = (float*)d_ws;
  float* qnT   = ws;
  float* knT   = qnT + (size_t)BN * FF * CC;
  float* ktot  = knT + (size_t)BN * FF * CC;
  float* kanch = ktot + BN * CC;
  float* s0    = kanch + BN * CC;

  // 1) x passthrough (first tuple element of the reference output)
  hipMemcpyAsync(out, x, (size_t)BN * CC * FF * sizeof(float),
                 hipMemcpyDeviceToDevice, stream);

  // 2) WMMA GEMMs + normalize (q and k via blockIdx.z)
  gemm_norm_kernel<<<dim3(FF / 16, BN, 2), 128, 0, stream>>>(x, Wq, Wk, qnT, knT);

  // 3) ktot (deterministic reduction)
  ktot_kernel<<<BN, 256, 0, stream>>>(knT, ktot);

  // 4) anchor k extraction
  anchor_kernel<<<BN, 256, 0, stream>>>(knT, anchors, kanch);

  // 5) initial scores
  scores0_kernel<<<dim3(FF / 32, BN), 256, 0, stream>>>(qnT, kanch, s0);

  // 6) sequential flood fill, one WGP per batch
  flood_kernel<<<BN, 1024, 0, stream>>>(qnT, knT, ktot, s0, adj, anchors, out_scores);
}